// DepthToSpace_15642270892754
// MI455X (gfx1250) — compile-verified
//
#include <hip/hip_runtime.h>

// 3D depth_to_space, block=2, channels_last.
//   in : (4, 32, 64, 64, 128) f32
//   out: (4, 64, 128, 128, 16) f32
//
// Pure permutation (0 FLOPs) -> HBM-bandwidth bound. 512 MB total traffic,
// ~22 us floor at 23.3 TB/s. Input working set (256 MB) > 192 MB L2 and is
// streamed once, so we use non-temporal cache hints (gfx1250 TH=NT) on both
// loads and stores, with full-width 128-bit vector memory ops.
//
// Structure: out[n,2z+dz,2y+dy,2x+dx,c] = in[n,z,y,x,(dz*4+dy*2+dx)*16+c].
// The dx=0/1 pair + c=0..15 form 32 contiguous floats in BOTH tensors
// (128-byte aligned on both sides), so the op is a pure b128 block copy with
// power-of-two index math (shifts/masks only).

typedef __attribute__((ext_vector_type(4))) float v4f;

__global__ __launch_bounds__(256) void d2s3d_b2_kernel(const v4f* __restrict__ in4,
                                                       v4f* __restrict__ out4,
                                                       unsigned total4) {
  unsigned q = blockIdx.x * 256u + threadIdx.x;   // output float4 index
  if (q >= total4) return;

  // Each 32-float (8 x float4) output block is one contiguous input block.
  unsigned w = q & 7u;          // float4 within the 32-float block
  unsigned m = q >> 3;          // block index, row-major over output blocks

  // m = ((n*64 + z')*128 + y')*64 + x   (all power-of-two dims)
  unsigned x  = m & 63u;          // input x (output x' pair = 2x, 2x+1)
  unsigned yp = (m >> 6) & 127u;  // output y' = 2y + dy
  unsigned zp = (m >> 13) & 63u;  // output z' = 2z + dz
  unsigned n  = m >> 19;          // batch

  unsigned j = ((zp & 1u) << 1) | (yp & 1u);      // dz*2 + dy
  // site = ((n*32 + z)*64 + y)*64 + x
  unsigned site = (((((n << 5) | (zp >> 1)) << 6) | (yp >> 1)) << 6) | x;
  // input float4 index: site*32 float4s per site, + j*8 + w within
  unsigned in_idx = (site << 5) | (j << 3) | w;

  v4f v = __builtin_nontemporal_load(in4 + in_idx);   // global_load_b128 th:NT
  __builtin_nontemporal_store(v, out4 + q);           // global_store_b128 th:NT
}

extern "C" void kernel_launch(void* const* d_in, const int* in_sizes, int n_in,
                              void* d_out, int out_size, void* d_ws, size_t ws_size,
                              hipStream_t stream) {
  (void)in_sizes; (void)n_in; (void)d_ws; (void)ws_size;
  const v4f* in4 = (const v4f*)d_in[0];
  v4f* out4 = (v4f*)d_out;

  unsigned total4 = (unsigned)(out_size / 4);         // 16,777,216 float4s
  unsigned blocks = (total4 + 255u) / 256u;           // 65,536 workgroups
  d2s3d_b2_kernel<<<blocks, 256, 0, stream>>>(in4, out4, total4);
}